// bwd_gnn_resid_45174466019872
// MI455X (gfx1250) — compile-verified
//
#include <hip/hip_runtime.h>
#include <math.h>

typedef float v2f __attribute__((ext_vector_type(2)));
typedef float v4f __attribute__((ext_vector_type(4)));
typedef float v8f __attribute__((ext_vector_type(8)));

#define L_   8
#define P_   16384
#define DEG_ 16
#define F_   32
#define H_   128

__device__ __forceinline__ v8f wmma4(v2f a, v2f b, v8f c) {
  return __builtin_amdgcn_wmma_f32_16x16x4_f32(false, a, false, b, (short)0, c, false, false);
}

// Wait for all of this wave's async (LDS-direct) loads to complete.
__device__ __forceinline__ void wait_async0() {
#if __has_builtin(__builtin_amdgcn_s_wait_asynccnt)
  __builtin_amdgcn_s_wait_asynccnt(0);
#else
  asm volatile("s_wait_asynccnt 0x0" ::: "memory");
#endif
}

// One wave-wide async 16B copy: global -> LDS (per-lane LDS dest keeps swizzle).
__device__ __forceinline__ void async_cp16(unsigned lds_byte_off, const float* src) {
  asm volatile("global_load_async_to_lds_b128 %0, %1, off"
               :: "v"(lds_byte_off), "v"(src)
               : "memory");
}

// ---------------- gather + sum over DEG predecessors ----------------
__global__ void gather_sum_kernel(const float* __restrict__ prev,
                                  const int* __restrict__ idx,
                                  float* __restrict__ out) {
  __shared__ int sidx[DEG_];
  int p = blockIdx.x;
  int t = threadIdx.x;                    // 0..127 == H columns
  if (t < DEG_) sidx[t] = idx[p * DEG_ + t];
  __syncthreads();
  float s = 0.f;
#pragma unroll
  for (int d = 0; d < DEG_; ++d)
    s += prev[(size_t)sidx[d] * H_ + t];
  out[(size_t)p * H_ + t] = s;
}

// ---------------- standalone embed (level 0 only) ----------------
__global__ void gemm_embed_kernel(const float* __restrict__ X,
                                  const float* __restrict__ W,
                                  const float* __restrict__ bias,
                                  float* __restrict__ out) {
  __shared__ float ldsW[H_ * 36];         // stride-36 padding: conflict-free
  int tid = threadIdx.x;
#pragma unroll
  for (int i = tid; i < H_ * (F_ / 4); i += 128) {
    int r = i >> 3, c4 = (i & 7) << 2;
    *(v4f*)&ldsW[r * 36 + c4] = *(const v4f*)(W + (size_t)r * F_ + c4);
  }
  __syncthreads();
  int lane = tid & 31, wv = tid >> 5;
  int row0 = blockIdx.x * 64 + wv * 16;
  int ml = lane & 15;
  int kb = (lane >> 4) << 1;              // 0 or 2
  const float* xr = X + (size_t)(row0 + ml) * F_;
  v8f acc[8] = {};
#pragma unroll
  for (int kk = 0; kk < F_; kk += 4) {
    v2f a = *(const v2f*)(xr + kk + kb);
#pragma unroll
    for (int n = 0; n < 8; ++n) {
      v2f b = *(const v2f*)&ldsW[(n * 16 + ml) * 36 + kk + kb];
      acc[n] = wmma4(a, b, acc[n]);
    }
  }
  int hi = (lane >> 4) << 3;
#pragma unroll
  for (int n = 0; n < 8; ++n) {
    int col = n * 16 + ml;
    float bv = bias[col];
#pragma unroll
    for (int i = 0; i < 8; ++i) {
      float v = acc[n][i] + bv;
      out[(size_t)(row0 + hi + i) * H_ + col] = tanhf(v);
    }
  }
}

// ---------------- fused-level helpers ----------------
// Async-stage a 128x128 weight matrix (row stride wld) into 64KB LDS, XOR-swizzled.
__device__ __forceinline__ void async_stage_w128(const float* __restrict__ W, int wld,
                                                 float* __restrict__ ldsW, int tid) {
  for (int i = tid; i < H_ * (H_ / 4); i += 128) {
    int r = i >> 5;
    int c4 = (i & 31) << 2;
    unsigned dst = (unsigned)(unsigned long long)&ldsW[(r << 7) + (c4 ^ ((r & 15) << 2))];
    async_cp16(dst, W + (size_t)r * wld + c4);
  }
}

// Async-stage the 128x32 embed weights into LDS with stride-36 padding.
__device__ __forceinline__ void async_stage_wemb(const float* __restrict__ W,
                                                 float* __restrict__ ldsW, int tid) {
  for (int i = tid; i < H_ * (F_ / 4); i += 128) {
    int r = i >> 3, c4 = (i & 7) << 2;
    unsigned dst = (unsigned)(unsigned long long)&ldsW[r * 36 + c4];
    async_cp16(dst, W + (size_t)r * F_ + c4);
  }
}

// K=128 GEMM accumulate: A from a wave-private swizzled 16x128 LDS act tile.
__device__ __forceinline__ void gemm_lds(const float* __restrict__ actT,
                                         const float* __restrict__ ldsW,
                                         v8f acc[8], int ml, int kb, int sw) {
#pragma unroll 4
  for (int kk = 0; kk < H_; kk += 4) {
    int o = (kk + kb) ^ sw;
    v2f a = *(const v2f*)&actT[(ml << 7) + o];
#pragma unroll
    for (int n = 0; n < 8; ++n) {
      v2f b = *(const v2f*)&ldsW[((n * 16 + ml) << 7) + o];
      acc[n] = wmma4(a, b, acc[n]);
    }
  }
}

__device__ __forceinline__ void bias_tanh(v8f acc[8], const float* __restrict__ bias, int ml) {
#pragma unroll
  for (int n = 0; n < 8; ++n) {
    float bv = bias[n * 16 + ml];
#pragma unroll
    for (int i = 0; i < 8; ++i) acc[n][i] = tanhf(acc[n][i] + bv);
  }
}

// Write D-layout registers into the wave's swizzled act tile.
__device__ __forceinline__ void write_act(float* __restrict__ actT, const v8f acc[8],
                                          int ml, int hi) {
#pragma unroll
  for (int n = 0; n < 8; ++n) {
    int col = n * 16 + ml;
#pragma unroll
    for (int i = 0; i < 8; ++i) {
      int r = hi + i;
      actT[(r << 7) + (col ^ (r << 2))] = acc[n][i];
    }
  }
}

// Read the wave's act tile back into D-layout registers (for residual saves).
__device__ __forceinline__ void read_act(const float* __restrict__ actT, v8f dst[8],
                                         int ml, int hi) {
#pragma unroll
  for (int n = 0; n < 8; ++n) {
    int col = n * 16 + ml;
#pragma unroll
    for (int i = 0; i < 8; ++i) {
      int r = hi + i;
      dst[n][i] = actT[(r << 7) + (col ^ (r << 2))];
    }
  }
}

// Load a 16x128 tile from global (coalesced b128) into the swizzled act tile.
__device__ __forceinline__ void load_act_global(const float* __restrict__ X, int row0,
                                                float* __restrict__ actT, int lane) {
  int c4 = lane << 2;
#pragma unroll
  for (int r = 0; r < 16; ++r) {
    v4f v = *(const v4f*)(X + (size_t)(row0 + r) * H_ + c4);
    *(v4f*)&actT[(r << 7) + (c4 ^ (r << 2))] = v;
  }
}

// ---------------- fused per-level kernel: embed + mp(9) + ne0 + ne(9) ----------------
// Weight staging is double-buffered with GLOBAL_LOAD_ASYNC_TO_LDS_B128:
//   stage j occupies buffer (j&1); at stage j we issue stage j+1 into buffer ((j+1)&1)
//   (safe: the barrier guarantees stage j-1's compute from that buffer is done),
//   then compute stage j while those async loads fly.
__global__ void __launch_bounds__(128)
level_fused_kernel(const float* __restrict__ msgs,       // [P,H] gathered messages
                   const float* __restrict__ nf,         // node_feats of level l+1 [P,32]
                   const float* __restrict__ W_embed, const float* __restrict__ b_embed,
                   const float* __restrict__ W_mp,    const float* __restrict__ b_mp,
                   const float* __restrict__ W_ne0,   const float* __restrict__ b_ne0,
                   const float* __restrict__ W_ne,    const float* __restrict__ b_ne,
                   float* __restrict__ out) {            // [P,H] level output
  __shared__ float ldsW0[H_ * H_];         // 64 KB weight buffer (even stages)
  __shared__ float ldsW1[H_ * H_];         // 64 KB weight buffer (odd stages)
  __shared__ float actA[4 * 16 * H_];      // 32 KB: per-wave activation tiles
  __shared__ float actB[4 * 16 * H_];      // 32 KB: per-wave embed tiles
  const int tid = threadIdx.x, lane = tid & 31, wv = tid >> 5;
  const int row0 = blockIdx.x * 64 + wv * 16;
  const int ml = lane & 15;
  const int kb = (lane >> 4) << 1;         // 0 or 2
  const int sw = ml << 2;
  const int hi = (lane >> 4) << 3;
  float* aT = actA + (wv << 11);
  float* bT = actB + (wv << 11);

  v8f acc[8];
  v8f sx[8];                               // single residual save slot

  // ---- prologue: stage 0 (embed W) async; overlap with msgs tile load ----
  async_stage_wemb(W_embed, ldsW0, tid);
  load_act_global(msgs, row0, aT, lane);
  read_act(aT, sx, ml, hi);                // x in D-layout for the layer-2 residual

  // ---- stage 0: embed level l+1 -> actB ----
  wait_async0();
  __syncthreads();
  async_stage_w128(W_mp, H_, ldsW1, tid);  // stage 1 (mp0) in flight
#pragma unroll
  for (int n = 0; n < 8; ++n) acc[n] = (v8f){};
  {
    const float* xr = nf + (size_t)(row0 + ml) * F_;
#pragma unroll
    for (int kk = 0; kk < F_; kk += 4) {
      v2f a = *(const v2f*)(xr + kk + kb);
#pragma unroll
      for (int n = 0; n < 8; ++n) {
        v2f b = *(const v2f*)&ldsW0[(n * 16 + ml) * 36 + kk + kb];
        acc[n] = wmma4(a, b, acc[n]);
      }
    }
  }
  bias_tanh(acc, b_embed, ml);
  write_act(bT, acc, ml, hi);

  // ---- stages 1..9: mp chain (layer i from buffer ((i+1)&1)) ----
#pragma unroll 1
  for (int i = 0; i < 9; ++i) {
    wait_async0();
    __syncthreads();
    float* nxt = (i & 1) ? ldsW1 : ldsW0;            // stage i+2 target
    float* cur = ((i + 1) & 1) ? ldsW1 : ldsW0;      // this layer's weights
    if (i < 8) async_stage_w128(W_mp + (size_t)(i + 1) * H_ * H_, H_, nxt, tid);
    else       async_stage_w128(W_ne0, 2 * H_, nxt, tid);   // stage 10 = ne0a
#pragma unroll
    for (int n = 0; n < 8; ++n) acc[n] = (v8f){};
    gemm_lds(aT, cur, acc, ml, kb, sw);
    bias_tanh(acc, b_mp + (size_t)i * H_, ml);
    if (i == 1 || i == 4 || i == 7) {      // next layer's input is h + saved
#pragma unroll
      for (int n = 0; n < 8; ++n) acc[n] += sx[n];
    }
    if (i == 2 || i == 5) {                // save r2 / r5
#pragma unroll
      for (int n = 0; n < 8; ++n) sx[n] = acc[n];
    }
    write_act(aT, acc, ml, hi);            // i==8 leaves redux in actA
  }

  // ---- stage 10: ne0a = emb @ W0[:, :128]^T (accumulators persist) ----
  wait_async0();
  __syncthreads();
  async_stage_w128(W_ne0 + H_, 2 * H_, ldsW1, tid);         // stage 11 = ne0b
#pragma unroll
  for (int n = 0; n < 8; ++n) acc[n] = (v8f){};
  gemm_lds(bT, ldsW0, acc, ml, kb, sw);

  // ---- stage 11: ne0b += redux @ W0[:, 128:]^T ----
  wait_async0();
  __syncthreads();
  async_stage_w128(W_ne, H_, ldsW0, tid);                   // stage 12 = ne[0]
  gemm_lds(aT, ldsW1, acc, ml, kb, sw);
  bias_tanh(acc, b_ne0, ml);
#pragma unroll
  for (int n = 0; n < 8; ++n) sx[n] = acc[n];               // save e0
  write_act(aT, acc, ml, hi);

  // ---- stages 12..20: ne chain (layer i from buffer (i&1)) ----
#pragma unroll 1
  for (int i = 0; i < 9; ++i) {
    wait_async0();
    __syncthreads();
    float* cur = (i & 1) ? ldsW1 : ldsW0;
    float* nxt = ((i + 1) & 1) ? ldsW1 : ldsW0;
    if (i < 8) async_stage_w128(W_ne + (size_t)(i + 1) * H_ * H_, H_, nxt, tid);
#pragma unroll
    for (int n = 0; n < 8; ++n) acc[n] = (v8f){};
    gemm_lds(aT, cur, acc, ml, kb, sw);
    bias_tanh(acc, b_ne + (size_t)i * H_, ml);
    if (i == 1 || i == 4 || i == 7) {
#pragma unroll
      for (int n = 0; n < 8; ++n) acc[n] += sx[n];
    }
    if (i == 2 || i == 5) {
#pragma unroll
      for (int n = 0; n < 8; ++n) sx[n] = acc[n];
    }
    if (i < 8) {
      write_act(aT, acc, ml, hi);
    } else {                               // final layer -> global out
#pragma unroll
      for (int n = 0; n < 8; ++n) {
        int col = n * 16 + ml;
#pragma unroll
        for (int j = 0; j < 8; ++j)
          out[(size_t)(row0 + hi + j) * H_ + col] = acc[n][j];
      }
    }
  }
}

extern "C" void kernel_launch(void* const* d_in, const int* in_sizes, int n_in,
                              void* d_out, int out_size, void* d_ws, size_t ws_size,
                              hipStream_t stream) {
  (void)in_sizes; (void)n_in; (void)out_size; (void)ws_size;
  const float* node_feats = (const float*)d_in[0];   // [N,32]
  const float* W_embed    = (const float*)d_in[1];   // [128,32]
  const float* b_embed    = (const float*)d_in[2];   // [128]
  const float* W_mp       = (const float*)d_in[3];   // [9,128,128]
  const float* b_mp       = (const float*)d_in[4];   // [9,128]
  const float* W_ne0      = (const float*)d_in[5];   // [128,256]
  const float* b_ne0      = (const float*)d_in[6];   // [128]
  const float* W_ne       = (const float*)d_in[7];   // [9,128,128]
  const float* b_ne       = (const float*)d_in[8];   // [9,128]
  const int*   pred_idx   = (const int*)d_in[9];     // [7,P,16]
  float* out = (float*)d_out;                        // [L*P,128]

  const size_t S = (size_t)P_ * H_;
  float* msgs = (float*)d_ws;                        // 8 MB scratch

  dim3 gg(P_ / 64), bb(128);

  // Level 0: embed directly into output rows [0, P)
  gemm_embed_kernel<<<gg, bb, 0, stream>>>(node_feats, W_embed, b_embed, out);

  for (int l = 0; l < L_ - 1; ++l) {
    gather_sum_kernel<<<P_, H_, 0, stream>>>(out + (size_t)l * S,
                                             pred_idx + (size_t)l * P_ * DEG_, msgs);
    level_fused_kernel<<<gg, bb, 0, stream>>>(msgs,
                                              node_feats + (size_t)(l + 1) * P_ * F_,
                                              W_embed, b_embed,
                                              W_mp, b_mp,
                                              W_ne0, b_ne0,
                                              W_ne, b_ne,
                                              out + (size_t)(l + 1) * S);
  }
}